// CrossSiloAggregator_79834852098715
// MI455X (gfx1250) — compile-verified
//
#include <hip/hip_runtime.h>

// Plain clang ext-vectors (HIP's float4 is a struct; __builtin_nontemporal_* needs real vectors)
typedef float v2f __attribute__((ext_vector_type(2)));
typedef float v8f __attribute__((ext_vector_type(8)));
typedef float f4v __attribute__((ext_vector_type(4)));

#define DIM 128
#define ROW_PAD 132            // 132 floats = 528B row pitch: 16B-aligned, bank-conflict-free A reads
#define WAVES_PER_BLOCK 4
#define ROWS_PER_WAVE 16
#define ROWS_PER_BLOCK (WAVES_PER_BLOCK * ROWS_PER_WAVE)

// ---------------------------------------------------------------------------
// Kernel 1: bulk clone of the 512MB table. Pure streaming: nontemporal 16B
// load/store so we don't thrash the 192MB L2 (working set is 1GB+).
// ---------------------------------------------------------------------------
__global__ __launch_bounds__(256) void table_copy_kernel(const f4v* __restrict__ src,
                                                         f4v* __restrict__ dst,
                                                         long n4) {
  long i = (long)blockIdx.x * blockDim.x + threadIdx.x;
  const long stride = (long)gridDim.x * blockDim.x;
  for (; i < n4; i += stride) {
    __builtin_nontemporal_store(__builtin_nontemporal_load(src + i), dst + i);
  }
}

// ---------------------------------------------------------------------------
// Kernel 2: fused gather -> WMMA dot -> sigmoid -> lerp -> scatter.
// One wave = one 16-row tile. V_WMMA_F32_16X16X4_F32 accumulates the
// [16 x 256] x [256 x 1] attention logits in exact f32.
//   A fragment (ISA 16x4 f32 layout): lane<16 -> {K0,K1}, lane>=16 -> {K2,K3},
//     row M = lane & 15.
//   B fragment: W chunk broadcast across all 16 columns (value depends only on
//     K, so only the K->lane/VGPR mapping matters; mirrored from A's layout).
//   D: lanes 0-15 hold rows 0-7 in acc[0..7]; lanes 16-31 hold rows 8-15.
// ---------------------------------------------------------------------------
__global__ __launch_bounds__(WAVES_PER_BLOCK * 32) void fuse_rows_kernel(
    const float* __restrict__ local_emb,   // [N_LOCAL, 128]
    const float* __restrict__ foreign,     // [M, 128]
    const int*   __restrict__ idx,         // [M]
    const float* __restrict__ W,           // [256]
    const float* __restrict__ bias_p,      // [1]
    float* __restrict__ out,               // [N_LOCAL, 128] (already cloned)
    int M) {
  __shared__ float sW[2 * DIM];
  __shared__ __align__(16) float sLoc[WAVES_PER_BLOCK][ROWS_PER_WAVE][ROW_PAD];
  __shared__ __align__(16) float sFor[WAVES_PER_BLOCK][ROWS_PER_WAVE][ROW_PAD];

  const int tid  = threadIdx.x;
  const int wave = tid >> 5;
  const int lane = tid & 31;
  const int half = lane >> 4;      // which 16-lane half of the wave
  const int m    = lane & 15;      // tile row for the A fragment
  const int R    = (blockIdx.x * WAVES_PER_BLOCK + wave) * ROWS_PER_WAVE;

  // Stage the attention weight vector once per block.
  for (int i = tid; i < 2 * DIM; i += WAVES_PER_BLOCK * 32) sW[i] = W[i];

  // Coalesced stage of 16 gathered local rows + 16 foreign rows into LDS.
  // (512B contiguous per row per table; data is reused by WMMA *and* the lerp.)
  for (int r = 0; r < ROWS_PER_WAVE; ++r) {
    int row = R + r;
    if (row >= M) row = M - 1;     // clamp: keeps loads valid, EXEC stays all-1s for WMMA
    const int g = idx[row];
    f4v lv = *(const f4v*)(local_emb + (long)g   * DIM + 4 * lane);
    f4v fv = *(const f4v*)(foreign   + (long)row * DIM + 4 * lane);
    *(f4v*)&sLoc[wave][r][4 * lane] = lv;
    *(f4v*)&sFor[wave][r][4 * lane] = fv;
  }
  __syncthreads();

  // 64 accumulating f32 WMMAs: K = 0..127 from local rows, 128..255 from foreign.
  v8f acc = {0.f, 0.f, 0.f, 0.f, 0.f, 0.f, 0.f, 0.f};
  const int kk = 2 * half;         // this half-wave's K offset within each K4 chunk
#pragma unroll 8
  for (int it = 0; it < 32; ++it) {
    const int k = 4 * it + kk;
    v2f a  = { sLoc[wave][m][k], sLoc[wave][m][k + 1] };
    v2f bb = { sW[k],            sW[k + 1] };
    acc = __builtin_amdgcn_wmma_f32_16x16x4_f32(false, a, false, bb,
                                                (short)0, acc, false, false);
  }
#pragma unroll 8
  for (int it = 0; it < 32; ++it) {
    const int k = 4 * it + kk;
    v2f a  = { sFor[wave][m][k],  sFor[wave][m][k + 1] };
    v2f bb = { sW[DIM + k],       sW[DIM + k + 1] };
    acc = __builtin_amdgcn_wmma_f32_16x16x4_f32(false, a, false, bb,
                                                (short)0, acc, false, false);
  }

  // Sigmoid on the per-lane logits (rows half*8 + r live in acc[r]).
  const float bias = bias_p[0];
  float w8[8];
#pragma unroll
  for (int r = 0; r < 8; ++r) {
    w8[r] = 1.0f / (1.0f + __expf(-(acc[r] + bias)));
  }

  // Lerp + scatter: broadcast row r's weight from lane 0 (rows 0-7) or lane 16
  // (rows 8-15), then every lane writes its 16B column slice (coalesced 512B).
#pragma unroll
  for (int r = 0; r < ROWS_PER_WAVE; ++r) {
    const int   row = R + r;
    const float wgt = __shfl(w8[r & 7], (r >> 3) << 4, 32);
    if (row < M) {
      const int g = idx[row];
      f4v lv = *(const f4v*)&sLoc[wave][r][4 * lane];
      f4v fv = *(const f4v*)&sFor[wave][r][4 * lane];
      f4v o  = wgt * lv + (1.0f - wgt) * fv;
      *(f4v*)(out + (long)g * DIM + 4 * lane) = o;
    }
  }
}

// ---------------------------------------------------------------------------
extern "C" void kernel_launch(void* const* d_in, const int* in_sizes, int n_in,
                              void* d_out, int out_size, void* d_ws, size_t ws_size,
                              hipStream_t stream) {
  const float* local_emb = (const float*)d_in[0];   // [N_LOCAL*128] f32
  const float* foreign   = (const float*)d_in[1];   // [M*128] f32
  const int*   idx       = (const int*)d_in[2];     // [M] i32
  const float* W         = (const float*)d_in[3];   // [256] f32
  const float* b         = (const float*)d_in[4];   // [1] f32
  float* out = (float*)d_out;

  const long n  = (long)in_sizes[0];                // N_LOCAL * 128
  const int  M  = in_sizes[2];
  const long n4 = n / 4;                            // DIM=128 -> divisible by 4

  // 1) clone the table (bandwidth-bound: ~44us of the ~52us roofline)
  long cb = (n4 + 255) / 256;
  int copyBlocks = (cb > 8192) ? 8192 : (int)cb;
  table_copy_kernel<<<copyBlocks, 256, 0, stream>>>((const f4v*)local_emb,
                                                    (f4v*)out, n4);

  // 2) fused boundary-row update, scattered over the clone (stream-ordered)
  const int blocks = (M + ROWS_PER_BLOCK - 1) / ROWS_PER_BLOCK;
  fuse_rows_kernel<<<blocks, WAVES_PER_BLOCK * 32, 0, stream>>>(
      local_emb, foreign, idx, W, b, out, M);
}